// MyViTAutoEncoder_26345329393924
// MI455X (gfx1250) — compile-verified
//
#include <hip/hip_runtime.h>
#include <math.h>

// ---------------------------------------------------------------------------
// MyViT autoencoder forward on gfx1250 (wave32, WMMA f32 16x16x4).
// Matmuls other than proj_kernel run on v_wmma_f32_16x16x4_f32. Transposed-output convention:
//   C-frag lane L holds column N=L%16, rows M = (L<16 ? 0..7 : 8..15).
//   A/B frags: lane L -> row/col L%16, K offset = 2*(L>>4) + vgpr.
// ---------------------------------------------------------------------------

typedef float v2f __attribute__((ext_vector_type(2)));
typedef float v8f __attribute__((ext_vector_type(8)));

#define WMMA_F32X4(a, b, c) \
  __builtin_amdgcn_wmma_f32_16x16x4_f32(false, (a), false, (b), (short)0, (c), false, false)

static __device__ inline v2f ld2(const float* p) { return *(const v2f*)p; }

// Convert one 4-row chunk (cl=0..3) of a C-layout fragment p[8] (P^T / H^T)
// into the B-fragment for the next WMMA (rows 4cl..4cl+3, cols = lane%16).
static __device__ inline v2f c2b(const float p[8], int cl, int hi) {
  int i0 = (4 * cl) & 7, i1 = (4 * cl + 1) & 7, i2 = (4 * cl + 2) & 7, i3 = (4 * cl + 3) & 7;
  float s0 = p[i0], s0p = p[i1], s1 = p[i2], s1p = p[i3];
  float t0  = __shfl_xor(s0, 16, 32);
  float t1  = __shfl_xor(s1, 16, 32);
  float t0p = __shfl_xor(s0p, 16, 32);
  float t1p = __shfl_xor(s1p, 16, 32);
  v2f b;
  if (cl < 2) { b.x = hi ? t1 : s0;  b.y = hi ? t1p : s0p; }   // source rows live in lanes 0-15
  else        { b.x = hi ? s1 : t0;  b.y = hi ? s1p : t0p; }   // source rows live in lanes 16-31
  return b;
}

// sinusoidal positional embedding (matches reference pos_emb)
static __device__ inline float pe_val(int p, int d) {
  // 10000^(-e) = exp2(-e * log2(10000))
  float e = (float)(d & ~1) * (1.0f / 32.0f);
  float ang = (float)p * exp2f(e * -13.287712379549449f);
  return (d & 1) ? cosf(ang) : sinf(ang);
}

// ---------------------------------------------------------------------------
// Patch slicing + linear embed + positional embedding.
//   x[b,p,d] = patch(b,p) @ W^T + bias + pe(p,d);  W: (32,96)
// One wave per 16 rows; X^T = W * patch^T via WMMA (2 M-tiles x 24 K-chunks).
// PN/IMGW are template constants so all /3 %3 gather offsets fold to literals.
// ---------------------------------------------------------------------------
template <int PN, int IMGW>
__global__ __launch_bounds__(32) void embed_kernel(
    const float* __restrict__ img, const float* __restrict__ W,
    const float* __restrict__ bias, float* __restrict__ xout) {
  const int lane = threadIdx.x, l16 = lane & 15, hi = lane >> 4, off = hi * 2;
  const int gr = blockIdx.x * 16 + l16;      // global row (b*PN + p)
  const int b = gr / PN, p = gr - b * PN;
  const long ib = (long)b * 32 * IMGW + 3 * p;
  v8f a0 = {}, a1 = {};
#pragma unroll
  for (int c = 0; c < 24; ++c) {
    // compile-time gather offsets for both lane halves
    const int o0 = ((4 * c + 0) / 3) * IMGW + (4 * c + 0) % 3;
    const int o1 = ((4 * c + 1) / 3) * IMGW + (4 * c + 1) % 3;
    const int o2 = ((4 * c + 2) / 3) * IMGW + (4 * c + 2) % 3;
    const int o3 = ((4 * c + 3) / 3) * IMGW + (4 * c + 3) % 3;
    v2f bf;
    bf.x = img[ib + (hi ? o2 : o0)];
    bf.y = img[ib + (hi ? o3 : o1)];
    v2f w0 = ld2(W + (long)l16 * 96 + 4 * c + off);
    v2f w1 = ld2(W + (long)(16 + l16) * 96 + 4 * c + off);
    a0 = WMMA_F32X4(w0, bf, a0);
    a1 = WMMA_F32X4(w1, bf, a1);
  }
  const long ob = (long)gr * 32;
#pragma unroll
  for (int m = 0; m < 8; ++m) {
    int d0 = 8 * hi + m, d1 = 16 + 8 * hi + m;
    xout[ob + d0] = a0[m] + bias[d0] + pe_val(p, d0);
    xout[ob + d1] = a1[m] + bias[d1] + pe_val(p, d1);
  }
}

// ---------------------------------------------------------------------------
// LayerNorm over last dim (32). One wave per row (8 rows / 256-thread block).
// ---------------------------------------------------------------------------
__global__ __launch_bounds__(256) void ln_kernel(
    const float* __restrict__ x, const float* __restrict__ g,
    const float* __restrict__ bt, float* __restrict__ y, int rows) {
  int row = blockIdx.x * 8 + (threadIdx.x >> 5);
  int d = threadIdx.x & 31;
  if (row >= rows) return;
  float v = x[(long)row * 32 + d];
  float s = v;
#pragma unroll
  for (int m = 16; m >= 1; m >>= 1) s += __shfl_xor(s, m, 32);
  float mean = s * (1.0f / 32.0f);
  float dv = v - mean;
  float s2 = dv * dv;
#pragma unroll
  for (int m = 16; m >= 1; m >>= 1) s2 += __shfl_xor(s2, m, 32);
  float var = s2 * (1.0f / 32.0f);
  y[(long)row * 32 + d] = dv * rsqrtf(var + 1e-5f) * g[d] + bt[d];
}

// ---------------------------------------------------------------------------
// Per-head projection: dst[r, h*8+e] = sum_d src[r, h*8+d] * W[h,d,e] + b[h,e]
// ---------------------------------------------------------------------------
__global__ __launch_bounds__(256) void proj_kernel(
    const float* __restrict__ src, const float* __restrict__ W,
    const float* __restrict__ bias, float* __restrict__ dst, int rows) {
  long tid = (long)blockIdx.x * blockDim.x + threadIdx.x;
  long r = tid >> 5;
  if (r >= rows) return;
  int c = (int)(tid & 31), h = c >> 3, e = c & 7;
  float acc = bias[h * 8 + e];
  const float* s = src + r * 32 + h * 8;
  const float* w = W + h * 64 + e;
#pragma unroll
  for (int d2 = 0; d2 < 8; ++d2) acc += s[d2] * w[d2 * 8];
  dst[r * 32 + c] = acc;
}

// ---------------------------------------------------------------------------
// Fused attention. One wave per (batch, head, 16-query tile).
// Computes S^T = K*Q^T and O^T = V^T*P^T so each lane owns one query column
// (softmax reductions = one xor-16 partner shuffle; flash rescale is a pure
// per-lane scalar multiply).
//  - CAUSAL=false: single-pass online softmax (6 WMMA / k-tile, K loaded once)
//  - CAUSAL=true:  two-pass, reproducing the reference's mask-AFTER-softmax
//                  (-1e9 fill) exactly.
// Accumulates result into xdst (residual add), disjoint per workgroup.
// ---------------------------------------------------------------------------
template <int SQ, int SKV, bool CAUSAL>
__global__ __launch_bounds__(32) void attn_kernel(
    const float* __restrict__ qbf, const float* __restrict__ kbf,
    const float* __restrict__ vbf, float* __restrict__ xdst) {
  const int lane = threadIdx.x, l16 = lane & 15, hi = lane >> 4, off = hi * 2;
  constexpr int nqt = SQ >> 4;
  int idx = blockIdx.x;
  const int b = idx / (4 * nqt); idx -= b * 4 * nqt;
  const int h = idx / nqt;
  const int q0 = (idx - h * nqt) << 4;
  const float scale = 0.35355339059327373f;   // 1/sqrt(8)

  const long qbase = (long)(b * SQ + q0 + l16) * 32 + h * 8;
  v2f qf0 = ld2(qbf + qbase + off);
  v2f qf1 = ld2(qbf + qbase + 4 + off);

  constexpr int nkt = SKV >> 4;
  const int q_glob = q0 + l16;
  v8f acc = {};
  float m_run = -3.4e38f, l_run = 0.0f;

  if (!CAUSAL) {
    // ---- single-pass flash attention ----
    for (int kt = 0; kt < nkt; ++kt) {
      const int k0 = kt << 4;
      const long kbase = (long)(b * SKV + k0 + l16) * 32 + h * 8;
      v2f kf0 = ld2(kbf + kbase + off);
      v2f kf1 = ld2(kbf + kbase + 4 + off);
      v8f st = {};
      st = WMMA_F32X4(kf0, qf0, st);
      st = WMMA_F32X4(kf1, qf1, st);
      float tm = -3.4e38f;
#pragma unroll
      for (int m = 0; m < 8; ++m) tm = fmaxf(tm, st[m] * scale);
      tm = fmaxf(tm, __shfl_xor(tm, 16, 32));
      float nm = fmaxf(m_run, tm);
      float alpha = __expf(m_run - nm);
      float p[8];
      float ts = 0.0f;
#pragma unroll
      for (int m = 0; m < 8; ++m) { p[m] = __expf(st[m] * scale - nm); ts += p[m]; }
      ts += __shfl_xor(ts, 16, 32);
      l_run = l_run * alpha + ts;
      m_run = nm;
#pragma unroll
      for (int m = 0; m < 8; ++m) acc[m] *= alpha;
#pragma unroll
      for (int cl = 0; cl < 4; ++cl) {
        v2f bf = c2b(p, cl, hi);
        int d = l16 & 7;
        long vb0 = (long)(b * SKV + k0 + 4 * cl + off) * 32 + h * 8 + d;
        float va0 = vbf[vb0];
        float va1 = vbf[vb0 + 32];
        if (l16 >= 8) { va0 = 0.0f; va1 = 0.0f; }
        v2f af; af.x = va0; af.y = va1;
        acc = WMMA_F32X4(af, bf, acc);
      }
    }
    const float inv_l = 1.0f / l_run;
#pragma unroll
    for (int m = 0; m < 8; ++m) acc[m] *= inv_l;
  } else {
    // ---- pass 1: per-query max & sum over ALL keys (softmax is unmasked) ----
    for (int kt = 0; kt < nkt; ++kt) {
      const long kbase = (long)(b * SKV + (kt << 4) + l16) * 32 + h * 8;
      v2f kf0 = ld2(kbf + kbase + off);
      v2f kf1 = ld2(kbf + kbase + 4 + off);
      v8f st = {};
      st = WMMA_F32X4(kf0, qf0, st);
      st = WMMA_F32X4(kf1, qf1, st);
      float tm = -3.4e38f;
#pragma unroll
      for (int m = 0; m < 8; ++m) tm = fmaxf(tm, st[m] * scale);
      tm = fmaxf(tm, __shfl_xor(tm, 16, 32));
      float ts = 0.0f;
#pragma unroll
      for (int m = 0; m < 8; ++m) ts += __expf(st[m] * scale - tm);
      ts += __shfl_xor(ts, 16, 32);
      float nm = fmaxf(m_run, tm);
      l_run = l_run * __expf(m_run - nm) + ts * __expf(tm - nm);
      m_run = nm;
    }
    const float inv_l = 1.0f / l_run;
    // ---- pass 2: P (with post-softmax -1e9 mask) @ V ----
    for (int kt = 0; kt < nkt; ++kt) {
      const int k0 = kt << 4;
      const long kbase = (long)(b * SKV + k0 + l16) * 32 + h * 8;
      v2f kf0 = ld2(kbf + kbase + off);
      v2f kf1 = ld2(kbf + kbase + 4 + off);
      v8f st = {};
      st = WMMA_F32X4(kf0, qf0, st);
      st = WMMA_F32X4(kf1, qf1, st);
      float p[8];
#pragma unroll
      for (int m = 0; m < 8; ++m) {
        float pv = __expf(st[m] * scale - m_run) * inv_l;
        if (k0 + hi * 8 + m > q_glob) pv = -1e9f;  // mask AFTER softmax
        p[m] = pv;
      }
#pragma unroll
      for (int cl = 0; cl < 4; ++cl) {
        v2f bf = c2b(p, cl, hi);
        int d = l16 & 7;
        long vb0 = (long)(b * SKV + k0 + 4 * cl + off) * 32 + h * 8 + d;
        float va0 = vbf[vb0];
        float va1 = vbf[vb0 + 32];
        if (l16 >= 8) { va0 = 0.0f; va1 = 0.0f; }
        v2f af; af.x = va0; af.y = va1;
        acc = WMMA_F32X4(af, bf, acc);
      }
    }
  }

  if (hi == 0) {
    long obase = (long)(b * SQ + q_glob) * 32 + h * 8;
#pragma unroll
    for (int m = 0; m < 8; ++m) xdst[obase + m] += acc[m];
  }
}

// ---------------------------------------------------------------------------
// MLP: xdst += gelu(n2 @ W1^T + b1) @ W2^T + b2. One wave per 16 rows.
// H^T = W1 * n2^T (8 M-tiles), then O^T = W2 * H^T via c2b shuffles.
// ---------------------------------------------------------------------------
__global__ __launch_bounds__(32) void mlp_kernel(
    const float* __restrict__ n2, const float* __restrict__ W1,
    const float* __restrict__ b1, const float* __restrict__ W2,
    const float* __restrict__ b2, float* __restrict__ xdst) {
  const int lane = threadIdx.x, l16 = lane & 15, hi = lane >> 4, off = hi * 2;
  const long r = (long)blockIdx.x * 16 + l16;
  v2f xf[8];
#pragma unroll
  for (int c = 0; c < 8; ++c) xf[c] = ld2(n2 + r * 32 + 4 * c + off);
  v8f hf[8];
#pragma unroll
  for (int t = 0; t < 8; ++t) {
    v8f a = {};
#pragma unroll
    for (int c = 0; c < 8; ++c) {
      v2f wf = ld2(W1 + (long)(16 * t + l16) * 32 + 4 * c + off);
      a = WMMA_F32X4(wf, xf[c], a);
    }
#pragma unroll
    for (int m = 0; m < 8; ++m) {
      float v = a[m] + b1[16 * t + 8 * hi + m];
      a[m] = 0.5f * v * (1.0f + erff(v * 0.70710678118654752f));  // exact GELU
    }
    hf[t] = a;
  }
#pragma unroll
  for (int t2 = 0; t2 < 2; ++t2) {
    v8f a = {};
#pragma unroll
    for (int c = 0; c < 32; ++c) {
      float p[8];
#pragma unroll
      for (int m = 0; m < 8; ++m) p[m] = hf[c >> 2][m];
      v2f bf = c2b(p, c & 3, hi);
      v2f wf = ld2(W2 + (long)(16 * t2 + l16) * 128 + 4 * c + off);
      a = WMMA_F32X4(wf, bf, a);
    }
#pragma unroll
    for (int m = 0; m < 8; ++m) {
      int dout = 16 * t2 + 8 * hi + m;
      xdst[r * 32 + dout] += a[m] + b2[dout];
    }
  }
}

// ---------------------------------------------------------------------------
// Final head + visualization scatter. sl = x @ mlp_W^T + mlp_b (96 outs),
// image[b,0,hrow,3p+cw] = sl[b,p,hrow*3+cw]; last patch (p==255) -> zeros.
// ---------------------------------------------------------------------------
__global__ __launch_bounds__(32) void final_kernel(
    const float* __restrict__ x, const float* __restrict__ W,
    const float* __restrict__ bias, float* __restrict__ out) {
  const int lane = threadIdx.x, l16 = lane & 15, hi = lane >> 4, off = hi * 2;
  const int gr = blockIdx.x * 16 + l16;  // row in 0..4095
  const int b = gr >> 8, p = gr & 255;
  v2f xf[8];
#pragma unroll
  for (int c = 0; c < 8; ++c) xf[c] = ld2(x + (long)gr * 32 + 4 * c + off);
#pragma unroll
  for (int t = 0; t < 6; ++t) {
    v8f a = {};
#pragma unroll
    for (int c = 0; c < 8; ++c) {
      v2f wf = ld2(W + (long)(16 * t + l16) * 32 + 4 * c + off);
      a = WMMA_F32X4(wf, xf[c], a);
    }
#pragma unroll
    for (int m = 0; m < 8; ++m) {
      int f = 16 * t + 8 * hi + m;           // unrolled: f/3, f%3 fold per variant
      float val = (p == 255) ? 0.0f : (a[m] + bias[f]);
      int hrow = f / 3, cw = f - hrow * 3;
      out[(long)b * 32 * 768 + (long)hrow * 768 + 3 * p + cw] = val;
    }
  }
}

// ---------------------------------------------------------------------------
// Host orchestration.
// Param leaf order = jax tree_leaves (sorted dict keys):
//  0:dec_W 1:dec_b
//  2+20i dec block i: {0:W1 1:W2 2:b1 3:b2 4:ca.Wk 5:ca.Wq 6:ca.Wv 7:ca.bk
//                      8:ca.bq 9:ca.bv 10:n1b 11:n1g 12:n2b 13:n2g
//                      14:sa.Wk 15:sa.Wq 16:sa.Wv 17:sa.bk 18:sa.bq 19:sa.bv}
//  42:enc_W 43:enc_b
//  44+14i enc block i: {0:W1 1:W2 2:b1 3:b2 4:n1b 5:n1g 6:n2b 7:n2g
//                       8:sa.Wk 9:sa.Wq 10:sa.Wv 11:sa.bk 12:sa.bq 13:sa.bv}
//  72:mlp_W 73:mlp_b
// ---------------------------------------------------------------------------
extern "C" void kernel_launch(void* const* d_in, const int* in_sizes, int n_in,
                              void* d_out, int out_size, void* d_ws, size_t ws_size,
                              hipStream_t stream) {
  (void)out_size; (void)ws_size;
  const int SRC_ELEMS = 16 * 32 * 3072;
  const float *SRC, *TGT; int ps;
  if (n_in >= 2 && in_sizes[0] == SRC_ELEMS) {
    SRC = (const float*)d_in[0]; TGT = (const float*)d_in[1]; ps = 2;      // [src, tgt, params...]
  } else {
    ps = 0; SRC = (const float*)d_in[n_in - 2]; TGT = (const float*)d_in[n_in - 1];  // [params..., src, tgt]
  }
  auto PP = [&](int i) { return (const float*)d_in[ps + i]; };

  float* ws   = (float*)d_ws;
  float* srcx = ws;                  // 16*1024*32
  float* nbuf = srcx + 524288;       // scratch for LN outputs (src-sized)
  float* qb   = nbuf + 524288;
  float* kb   = qb + 524288;
  float* vb   = kb + 524288;
  float* tgtx = vb + 524288;         // 16*256*32

  // ---------------- encoder ----------------
  embed_kernel<1024, 3072><<<1024, 32, 0, stream>>>(SRC, PP(42), PP(43), srcx);
  for (int eb = 0; eb < 2; ++eb) {
    int B0 = 44 + eb * 14;
    ln_kernel<<<2048, 256, 0, stream>>>(srcx, PP(B0 + 5), PP(B0 + 4), nbuf, 16384);
    proj_kernel<<<2048, 256, 0, stream>>>(nbuf, PP(B0 + 9),  PP(B0 + 12), qb, 16384);
    proj_kernel<<<2048, 256, 0, stream>>>(nbuf, PP(B0 + 8),  PP(B0 + 11), kb, 16384);
    proj_kernel<<<2048, 256, 0, stream>>>(nbuf, PP(B0 + 10), PP(B0 + 13), vb, 16384);
    attn_kernel<1024, 1024, false><<<4096, 32, 0, stream>>>(qb, kb, vb, srcx);
    ln_kernel<<<2048, 256, 0, stream>>>(srcx, PP(B0 + 7), PP(B0 + 6), nbuf, 16384);
    mlp_kernel<<<1024, 32, 0, stream>>>(nbuf, PP(B0 + 0), PP(B0 + 2), PP(B0 + 1), PP(B0 + 3), srcx);
  }
  // ---------------- decoder ----------------
  embed_kernel<256, 768><<<256, 32, 0, stream>>>(TGT, PP(0), PP(1), tgtx);
  for (int db = 0; db < 2; ++db) {
    int B0 = 2 + db * 20;
    // masked self-attention (two-pass, post-softmax mask)
    ln_kernel<<<512, 256, 0, stream>>>(tgtx, PP(B0 + 11), PP(B0 + 10), nbuf, 4096);
    proj_kernel<<<512, 256, 0, stream>>>(nbuf, PP(B0 + 15), PP(B0 + 18), qb, 4096);
    proj_kernel<<<512, 256, 0, stream>>>(nbuf, PP(B0 + 14), PP(B0 + 17), kb, 4096);
    proj_kernel<<<512, 256, 0, stream>>>(nbuf, PP(B0 + 16), PP(B0 + 19), vb, 4096);
    attn_kernel<256, 256, true><<<1024, 32, 0, stream>>>(qb, kb, vb, tgtx);
    // cross-attention: q from out (unnormalized), k/v from LN(enc_out) w/ dec n1 params
    ln_kernel<<<2048, 256, 0, stream>>>(srcx, PP(B0 + 11), PP(B0 + 10), nbuf, 16384);
    proj_kernel<<<512, 256, 0, stream>>>(tgtx, PP(B0 + 5), PP(B0 + 8), qb, 4096);
    proj_kernel<<<2048, 256, 0, stream>>>(nbuf, PP(B0 + 4), PP(B0 + 7), kb, 16384);
    proj_kernel<<<2048, 256, 0, stream>>>(nbuf, PP(B0 + 6), PP(B0 + 9), vb, 16384);
    attn_kernel<256, 1024, false><<<1024, 32, 0, stream>>>(qb, kb, vb, tgtx);
    // MLP
    ln_kernel<<<512, 256, 0, stream>>>(tgtx, PP(B0 + 13), PP(B0 + 12), nbuf, 4096);
    mlp_kernel<<<256, 32, 0, stream>>>(nbuf, PP(B0 + 0), PP(B0 + 2), PP(B0 + 1), PP(B0 + 3), tgtx);
  }
  final_kernel<<<256, 32, 0, stream>>>(tgtx, PP(72), PP(73), (float*)d_out);
}